// Sampler_48430051230166
// MI455X (gfx1250) — compile-verified
//
#include <hip/hip_runtime.h>
#include <math.h>

// ---------------- problem constants ----------------
#define NSEQ   512
#define VOCAB  128000
#define PLEN   1024
#define OLEN   256
#define TB     1024                 // threads per block
#define VEC_ITERS 31                // 31*1024 float4 = 126976 floats
#define REM_BASE  126976            // + tid covers the last 1024 floats
#define HBINS  65536
#define N2     2048                 // candidate / token-table capacity

// shared memory layout (bytes)
#define SM_HIST  0
#define SM_TOK   (HBINS * 4)                 // tokk[2048] u32
#define SM_VAL   (SM_TOK  + N2 * 4)          // tval[2048] f32
#define SM_CANDO (SM_VAL  + N2 * 4)          // cand[2048] u32 (also bsum scratch)
#define SM_EX    (SM_CANDO + N2 * 4)         // exs[2048] f32
#define SM_SCAL  (SM_EX   + N2 * 4)          // 16 scalar slots
#define SM_TOTAL (SM_SCAL + 64)              // = 294976 B  (< 320KB WGP LDS)

typedef float v4f __attribute__((ext_vector_type(4)));

// ---------------- order-preserving float<->key ----------------
__device__ __forceinline__ unsigned f2key(float f) {
    unsigned u = __float_as_uint(f);
    return (u & 0x80000000u) ? ~u : (u | 0x80000000u);
}
__device__ __forceinline__ float key2f(unsigned k) {
    unsigned u = (k & 0x80000000u) ? (k & 0x7fffffffu) : ~k;
    return __uint_as_float(u);
}

// ---------------- async global->LDS path (codegen-verified on this toolchain) ----------------
#if defined(__has_builtin)
#if __has_builtin(__builtin_amdgcn_global_load_async_to_lds_b128) && \
    __has_builtin(__builtin_amdgcn_s_wait_asynccnt)
#define ASYNC_PATH 1
#endif
#endif
#ifndef ASYNC_PATH
#define ASYNC_PATH 0
#endif

#if ASYNC_PATH
typedef int v4i __attribute__((ext_vector_type(4)));
typedef __attribute__((address_space(1))) v4i* gv4i_p;   // global pointer to v4i
typedef __attribute__((address_space(3))) v4i* lv4i_p;   // LDS pointer to v4i
#endif

// ---------------- penalty formula ----------------
__device__ __forceinline__ float penalize(float l, int cnt, float rp, float fp, float pp) {
    l = (l > 0.f) ? (l / rp) : (l * rp);     // token is "seen" by construction
    l -= fp * (float)cnt;
    if (cnt > 0) l -= pp;
    return l;
}

// ---------------- branchless lower-bound over the sorted 2048-entry token table ----------------
__device__ __forceinline__ float lut_lookup(const unsigned* tokk, const float* tval,
                                            unsigned j, float raw) {
    int lo = 0, hi = N2 - 1;
    #pragma unroll
    for (int it = 0; it < 11; ++it) {
        int mid = (lo + hi) >> 1;
        bool c = tokk[mid] < j;
        lo = c ? (mid + 1) : lo;
        hi = c ? hi : mid;
    }
    return (tokk[lo] == j) ? tval[lo] : raw;
}

// ---------------- descending rank select over 65536-bin LDS histogram ----------------
__device__ void select_desc(unsigned* hist, unsigned* bsum, unsigned* scal,
                            unsigned k, int outBinIdx) {
    const int t = threadIdx.x;
    unsigned s = 0;
    const int base = t * 64;
    #pragma unroll 8
    for (int q = 0; q < 64; ++q) s += hist[base + q];
    bsum[t] = s;
    __syncthreads();
    if (t == 0) {
        unsigned cum = 0;
        int b = 1023;
        for (; b >= 0; --b) {
            unsigned ns = cum + bsum[b];
            if (ns >= k) break;
            cum = ns;
        }
        if (b < 0) b = 0;
        unsigned rem = k - cum;
        unsigned c2 = 0;
        int q = 63;
        for (; q >= 0; --q) {
            unsigned ns = c2 + hist[b * 64 + q];
            if (ns >= rem) break;
            c2 = ns;
        }
        if (q < 0) q = 0;
        scal[outBinIdx] = (unsigned)(b * 64 + q);
    }
    __syncthreads();
}

// ---------------- fused sampler: penalties + top-k + top-p + min-p + softmax ----------------
__global__ void __launch_bounds__(TB, 1)
sampler_fused(const float* __restrict__ logits,
              const int*   __restrict__ prompt,
              const int*   __restrict__ outtok,
              const float* __restrict__ pres,
              const float* __restrict__ freq,
              const float* __restrict__ rep,
              const float* __restrict__ top_p,
              const float* __restrict__ min_p,
              const int*   __restrict__ top_k,
              float* __restrict__ dout) {
    extern __shared__ unsigned char smem[];
    unsigned* hist = (unsigned*)(smem + SM_HIST);
    unsigned* tokk = (unsigned*)(smem + SM_TOK);
    float*    tval = (float*)(smem + SM_VAL);
    unsigned* cand = (unsigned*)(smem + SM_CANDO);
    float*    exs  = (float*)(smem + SM_EX);
    unsigned* scal = (unsigned*)(smem + SM_SCAL);

    const int row = blockIdx.x;
    const int t = threadIdx.x;
    const size_t rbase = (size_t)row * VOCAB;
    const float* rowl = logits + rbase;
    float* rowo = dout + rbase;

    // ---- init hist/scalars + async-stream token lists into LDS ----
    for (int q = t; q < HBINS; q += TB) hist[q] = 0u;
    if (t < 16) scal[t] = 0u;
#if ASYNC_PATH
    if (t < PLEN / 4)
        __builtin_amdgcn_global_load_async_to_lds_b128(
            (gv4i_p)(prompt + row * PLEN + 4 * t), (lv4i_p)&tokk[4 * t], 0, 0);
    if (t < OLEN / 4)
        __builtin_amdgcn_global_load_async_to_lds_b128(
            (gv4i_p)(outtok + row * OLEN + 4 * t), (lv4i_p)&tokk[PLEN + 4 * t], 0, 0);
    __builtin_amdgcn_s_wait_asynccnt(0);
#else
    tokk[t] = (unsigned)prompt[row * PLEN + t];
    if (t < OLEN) tokk[PLEN + t] = (unsigned)outtok[row * OLEN + t];
#endif
    __syncthreads();

    // ---- build (token, penalized value) table; pad to 2048 with sentinel ----
    const float rp = rep[row], fp = freq[row], pp = pres[row];
    {
        unsigned tok0 = tokk[t];
        int c0 = 0;
        #pragma unroll 8
        for (int j = 0; j < OLEN; ++j) c0 += (tokk[PLEN + j] == tok0);
        tval[t] = penalize(rowl[tok0], c0, rp, fp, pp);
        if (t < OLEN) {
            unsigned tok1 = tokk[PLEN + t];
            int c1 = 0;
            #pragma unroll 8
            for (int j = 0; j < OLEN; ++j) c1 += (tokk[PLEN + j] == tok1);
            tval[PLEN + t] = penalize(rowl[tok1], c1, rp, fp, pp);
        } else {
            tokk[PLEN + t] = 0xFFFFFFFFu;    // pad slots 1280..2047
            tval[PLEN + t] = 0.f;
        }
    }
    __syncthreads();

    // ---- bitonic sort token table ascending by token id (N=2048, 2 elems/thread) ----
    for (unsigned sz = 2; sz <= N2; sz <<= 1) {
        for (unsigned j = sz >> 1; j > 0; j >>= 1) {
            #pragma unroll
            for (int r_ = 0; r_ < 2; ++r_) {
                unsigned i = (unsigned)t + (unsigned)r_ * 1024u;
                unsigned ixj = i ^ j;
                if (ixj > i) {
                    unsigned a = tokk[i], b = tokk[ixj];
                    bool ascBlk = ((i & sz) == 0);
                    if (ascBlk ? (a > b) : (a < b)) {
                        tokk[i] = b; tokk[ixj] = a;
                        float fa = tval[i], fb = tval[ixj];
                        tval[i] = fb; tval[ixj] = fa;
                    }
                }
            }
            __syncthreads();
        }
    }

    // ---- pass 1: read raw logits, penalize via table, hist + max (no global write) ----
    unsigned lmax = 0u;
    for (int v = 0; v < VEC_ITERS; ++v) {
        const int idx4 = v * TB + t;
        float4 x = ((const float4*)rowl)[idx4];
        const unsigned jb = (unsigned)idx4 * 4u;
        x.x = lut_lookup(tokk, tval, jb + 0u, x.x);
        x.y = lut_lookup(tokk, tval, jb + 1u, x.y);
        x.z = lut_lookup(tokk, tval, jb + 2u, x.z);
        x.w = lut_lookup(tokk, tval, jb + 3u, x.w);
        unsigned k0 = f2key(x.x), k1 = f2key(x.y), k2 = f2key(x.z), k3 = f2key(x.w);
        lmax = max(max(lmax, k0), max(k1, max(k2, k3)));
        atomicAdd(&hist[k0 >> 16], 1u);
        atomicAdd(&hist[k1 >> 16], 1u);
        atomicAdd(&hist[k2 >> 16], 1u);
        atomicAdd(&hist[k3 >> 16], 1u);
    }
    {
        const int idx = REM_BASE + t;
        float x = lut_lookup(tokk, tval, (unsigned)idx, rowl[idx]);
        unsigned kx = f2key(x);
        lmax = max(lmax, kx);
        atomicAdd(&hist[kx >> 16], 1u);
    }
    atomicMax(&scal[0], lmax);
    __syncthreads();

    // ---- locate 16-bit bucket of the k-th largest ----
    int kk = top_k[row];
    if (kk < 1) kk = 1;
    if (kk > 1000) kk = 1000;
    select_desc(hist, cand, scal, (unsigned)kk, 2);
    const unsigned bucketBase = scal[2] << 16;
    __syncthreads();

    // ---- gather all elements >= bucketBase (L2-hot re-read of read-only logits) ----
    cand[t] = 0u; cand[t + 1024] = 0u;
    if (t == 0) scal[1] = 0u;
    __syncthreads();
    for (int v = 0; v < VEC_ITERS; ++v) {
        const int idx4 = v * TB + t;
        float4 x = ((const float4*)rowl)[idx4];
        const unsigned jb = (unsigned)idx4 * 4u;
        unsigned kA[4];
        kA[0] = f2key(lut_lookup(tokk, tval, jb + 0u, x.x));
        kA[1] = f2key(lut_lookup(tokk, tval, jb + 1u, x.y));
        kA[2] = f2key(lut_lookup(tokk, tval, jb + 2u, x.z));
        kA[3] = f2key(lut_lookup(tokk, tval, jb + 3u, x.w));
        #pragma unroll
        for (int e = 0; e < 4; ++e)
            if (kA[e] >= bucketBase) {
                unsigned p = atomicAdd(&scal[1], 1u);
                if (p < N2) cand[p] = kA[e];
            }
    }
    {
        const int idx = REM_BASE + t;
        unsigned kx = f2key(lut_lookup(tokk, tval, (unsigned)idx, rowl[idx]));
        if (kx >= bucketBase) {
            unsigned p = atomicAdd(&scal[1], 1u);
            if (p < N2) cand[p] = kx;
        }
    }
    __syncthreads();
    unsigned ncand = scal[1];
    if (ncand > N2) ncand = N2;
    if (ncand < 1) ncand = 1;
    if ((unsigned)kk > ncand) kk = (int)ncand;

    // ---- bitonic sort candidates descending (keys only) ----
    for (unsigned sz = 2; sz <= N2; sz <<= 1) {
        for (unsigned j = sz >> 1; j > 0; j >>= 1) {
            #pragma unroll
            for (int r_ = 0; r_ < 2; ++r_) {
                unsigned i = (unsigned)t + (unsigned)r_ * 1024u;
                unsigned ixj = i ^ j;
                if (ixj > i) {
                    unsigned a = cand[i], b = cand[ixj];
                    bool descBlk = ((i & sz) == 0);
                    if (descBlk ? (a < b) : (a > b)) { cand[i] = b; cand[ixj] = a; }
                }
            }
            __syncthreads();
        }
    }

    // ---- exps over the top-k prefix (ties included), inclusive scan over 2048 ----
    const unsigned kthKey = cand[kk - 1];
    const float maxv = key2f(scal[0]);
    unsigned c0 = cand[t], c1 = cand[t + 1024];
    float e0 = (t < (int)ncand && c0 >= kthKey) ? __expf(key2f(c0) - maxv) : 0.f;
    float e1 = ((t + 1024) < (int)ncand && c1 >= kthKey) ? __expf(key2f(c1) - maxv) : 0.f;
    exs[t] = e0; exs[t + 1024] = e1;
    __syncthreads();
    for (int off = 1; off < N2; off <<= 1) {
        float a0 = (t >= off) ? exs[t - off] : 0.f;
        const int i1 = t + 1024;
        float a1 = (i1 >= off) ? exs[i1 - off] : 0.f;
        __syncthreads();
        exs[t] += a0; exs[i1] += a1;
        __syncthreads();
    }
    const float Zk  = exs[N2 - 1];            // exps are 0 outside the top-k set
    const float pZk = top_p[row] * Zk;
    const float mp  = min_p[row];
    // kept iff in top-k set, strict-above-sum < p*Zk (top-p), e >= min_p (min-p)
    const bool k0b = (e0 > 0.f) && ((exs[t] - e0) < pZk) && (e0 >= mp);
    const bool k1b = (e1 > 0.f) && ((exs[t + 1024] - e1) < pZk) && (e1 >= mp);

    if (t == 0) scal[5] = 0u;
    __syncthreads();
    {
        int add = (int)k0b + (int)k1b;
        if (add) atomicAdd(&scal[5], (unsigned)add);
    }
    __syncthreads();
    unsigned m = scal[5];
    if (m < 1) m = 1;
    if (m > ncand) m = ncand;
    const unsigned Tkey = cand[m - 1];        // kept set is a descending prefix
    const float Zf = exs[m - 1];
    const float invZ = 1.0f / Zf;

    // ---- final pass: re-read logits, re-penalize, write probabilities (NT store) ----
    for (int v = 0; v < VEC_ITERS; ++v) {
        const int idx4 = v * TB + t;
        float4 x = ((const float4*)rowl)[idx4];
        const unsigned jb = (unsigned)idx4 * 4u;
        x.x = lut_lookup(tokk, tval, jb + 0u, x.x);
        x.y = lut_lookup(tokk, tval, jb + 1u, x.y);
        x.z = lut_lookup(tokk, tval, jb + 2u, x.z);
        x.w = lut_lookup(tokk, tval, jb + 3u, x.w);
        v4f o;
        o.x = (f2key(x.x) >= Tkey) ? __expf(x.x - maxv) * invZ : 0.f;
        o.y = (f2key(x.y) >= Tkey) ? __expf(x.y - maxv) * invZ : 0.f;
        o.z = (f2key(x.z) >= Tkey) ? __expf(x.z - maxv) * invZ : 0.f;
        o.w = (f2key(x.w) >= Tkey) ? __expf(x.w - maxv) * invZ : 0.f;
        __builtin_nontemporal_store(o, (v4f*)(rowo + (size_t)idx4 * 4u));
    }
    {
        const int idx = REM_BASE + t;
        float x = lut_lookup(tokk, tval, (unsigned)idx, rowl[idx]);
        float o = (f2key(x) >= Tkey) ? __expf(x - maxv) * invZ : 0.f;
        __builtin_nontemporal_store(o, rowo + idx);
    }
}

// ---------------- host launcher ----------------
extern "C" void kernel_launch(void* const* d_in, const int* in_sizes, int n_in,
                              void* d_out, int out_size, void* d_ws, size_t ws_size,
                              hipStream_t stream) {
    (void)in_sizes; (void)n_in; (void)out_size; (void)d_ws; (void)ws_size;
    const float* logits = (const float*)d_in[0];
    const float* pres   = (const float*)d_in[1];
    const float* freq   = (const float*)d_in[2];
    const float* rep    = (const float*)d_in[3];
    const float* topp   = (const float*)d_in[4];
    const float* minp   = (const float*)d_in[5];
    const int*   prompt = (const int*)d_in[6];
    const int*   outtok = (const int*)d_in[7];
    const int*   topk   = (const int*)d_in[8];
    float* out = (float*)d_out;

    // Allow the large (~288KB) dynamic LDS allocation (gfx1250: 320KB/WGP).
    (void)hipFuncSetAttribute((const void*)sampler_fused,
                              hipFuncAttributeMaxDynamicSharedMemorySize, SM_TOTAL);

    sampler_fused<<<NSEQ, TB, SM_TOTAL, stream>>>(logits, prompt, outtok,
                                                  pres, freq, rep,
                                                  topp, minp, topk, out);
}